// PointNetSAModule_34754875359388
// MI455X (gfx1250) — compile-verified
//
#include <hip/hip_runtime.h>

#define BATCH      4
#define NPTS       8192
#define CFEAT      32
#define CIN        35
#define M_CENTERS  2048
#define KNB        32
#define RAD2       0.04f

typedef __attribute__((ext_vector_type(2))) float v2f;
typedef __attribute__((ext_vector_type(8))) float v8f;

// ---------------------------------------------------------------------------
// D = A(16x4 f32) * B(4x16 f32) + C(16x16 f32)   — native f32 WMMA on CDNA5
// ---------------------------------------------------------------------------
__device__ __forceinline__ v8f wmma_f32(v2f a, v2f b, v8f c) {
    return __builtin_amdgcn_wmma_f32_16x16x4_f32(
        /*neg_a=*/false, a, /*neg_b=*/false, b,
        /*c_mod=*/(short)0, c, /*reuse_a=*/false, /*reuse_b=*/false);
}

// ---------------------------------------------------------------------------
// Kernel 1: furthest point sampling. One block per batch, 1024 threads,
// 8 points per thread held in registers; argmax via wave shuffles + LDS.
// Writes centers (B,3,M) into the tail of d_out.
// ---------------------------------------------------------------------------
__global__ __launch_bounds__(1024) void fps_kernel(const float* __restrict__ coords,
                                                   float* __restrict__ centers) {
    const int b   = blockIdx.x;
    const int tid = threadIdx.x;
    const int lane = tid & 31;
    const int wid  = tid >> 5;
    const float* cb = coords + (size_t)b * 3 * NPTS;

    float px[8], py[8], pz[8], dst[8];
#pragma unroll
    for (int j = 0; j < 8; ++j) {
        int p = j * 1024 + tid;
        px[j] = cb[p];
        py[j] = cb[NPTS + p];
        pz[j] = cb[2 * NPTS + p];
        dst[j] = __builtin_inff();
    }

    __shared__ float s_val[32];
    __shared__ int   s_idx[32];
    __shared__ float s_cent[3];
    __shared__ int   s_far;

    int far = 0;
    for (int m = 0; m < M_CENTERS; ++m) {
        if (tid == 0) {
            float x = cb[far], y = cb[NPTS + far], z = cb[2 * NPTS + far];
            s_cent[0] = x; s_cent[1] = y; s_cent[2] = z;
            centers[(size_t)b * 3 * M_CENTERS + m]                 = x;
            centers[(size_t)b * 3 * M_CENTERS + M_CENTERS + m]     = y;
            centers[(size_t)b * 3 * M_CENTERS + 2 * M_CENTERS + m] = z;
        }
        __syncthreads();
        const float cx = s_cent[0], cy = s_cent[1], cz = s_cent[2];

        float bv = -1.0f; int bi = 0x7fffffff;
#pragma unroll
        for (int j = 0; j < 8; ++j) {
            float dx = px[j] - cx, dy = py[j] - cy, dz = pz[j] - cz;
            float dd = dx * dx + dy * dy + dz * dz;
            dd = fminf(dst[j], dd);
            dst[j] = dd;
            int p = j * 1024 + tid;
            if (dd > bv || (dd == bv && p < bi)) { bv = dd; bi = p; }
        }
#pragma unroll
        for (int s = 16; s > 0; s >>= 1) {
            float ov = __shfl_xor(bv, s);
            int   oi = __shfl_xor(bi, s);
            if (ov > bv || (ov == bv && oi < bi)) { bv = ov; bi = oi; }
        }
        if (lane == 0) { s_val[wid] = bv; s_idx[wid] = bi; }
        __syncthreads();
        if (wid == 0) {
            bv = s_val[lane]; bi = s_idx[lane];
#pragma unroll
            for (int s = 16; s > 0; s >>= 1) {
                float ov = __shfl_xor(bv, s);
                int   oi = __shfl_xor(bi, s);
                if (ov > bv || (ov == bv && oi < bi)) { bv = ov; bi = oi; }
            }
            if (lane == 0) s_far = bi;
        }
        __syncthreads();
        far = s_far;
    }
}

// ---------------------------------------------------------------------------
// Kernel 2: ball query. One wave32 per center; ballot-based ordered selection
// of the first K in-radius indices (matches sort(where(d2<r2,idx,N))[:k]).
// ---------------------------------------------------------------------------
__global__ __launch_bounds__(256) void ballquery_kernel(const float* __restrict__ coords,
                                                        const float* __restrict__ centers,
                                                        int* __restrict__ nbidx) {
    const int lane = threadIdx.x & 31;
    const int gw   = blockIdx.x * (blockDim.x >> 5) + (threadIdx.x >> 5);
    const int b    = gw / M_CENTERS;
    const int m    = gw % M_CENTERS;
    const float* cb = coords + (size_t)b * 3 * NPTS;

    const float cx = centers[(size_t)b * 3 * M_CENTERS + m];
    const float cy = centers[(size_t)b * 3 * M_CENTERS + M_CENTERS + m];
    const float cz = centers[(size_t)b * 3 * M_CENTERS + 2 * M_CENTERS + m];

    int myidx = -1;
    int count = 0;
    for (int base = 0; base < NPTS; base += 32) {
        const int p = base + lane;
        float dx = cb[p] - cx;
        float dy = cb[NPTS + p] - cy;
        float dz = cb[2 * NPTS + p] - cz;
        float d2 = dx * dx + dy * dy + dz * dz;
        unsigned mask = (unsigned)__ballot(d2 < RAD2);
        int c = __popc(mask);
        if (myidx < 0 && lane >= count && lane < count + c) {
            int t = lane - count;
            unsigned mm = mask;
            for (int i = 0; i < t; ++i) mm &= mm - 1;  // drop t lowest set bits
            myidx = base + (__ffs(mm) - 1);
        }
        count += c;
        if (count >= KNB) break;  // uniform across the wave
    }
    int first = __shfl(myidx, 0);
    if (myidx < 0) myidx = (first < 0) ? 0 : first;
    nbidx[(size_t)gw * KNB + lane] = myidx;
}

// ---------------------------------------------------------------------------
// Kernel 3: gather + 3-layer pointwise MLP (f32 WMMA) + max over neighbors.
// One wave per center; LDS ping-pong buffers hold the C x 32 activations.
// ---------------------------------------------------------------------------
template <int CDIM, int CPAD, int ODIM>
__device__ __forceinline__ void mlp_layer(const float* __restrict__ W,
                                          const float* __restrict__ bias,
                                          const float* xin, float* xout, int lane) {
    const int half = lane >> 4;   // 0: K/M rows 0..; 1: +2 K-offset / +8 M-offset
    const int l    = lane & 15;
    for (int mt = 0; mt < ODIM / 16; ++mt) {
        for (int nt = 0; nt < 2; ++nt) {
            v8f acc = {0.f, 0.f, 0.f, 0.f, 0.f, 0.f, 0.f, 0.f};
            for (int k0 = 0; k0 < CPAD; k0 += 4) {
                const int kk = k0 + half * 2;
                v2f a, bb;
                a.x = (kk     < CDIM) ? W[(mt * 16 + l) * CDIM + kk]     : 0.f;
                a.y = (kk + 1 < CDIM) ? W[(mt * 16 + l) * CDIM + kk + 1] : 0.f;
                bb.x = xin[kk * KNB + nt * 16 + l];        // padded rows are zero
                bb.y = xin[(kk + 1) * KNB + nt * 16 + l];
                acc = wmma_f32(a, bb, acc);
            }
#pragma unroll
            for (int r = 0; r < 8; ++r) {
                const int row = mt * 16 + half * 8 + r;    // C/D layout rule
                const int col = nt * 16 + l;
                float v = acc[r] + bias[row];
                xout[row * KNB + col] = v > 0.f ? v : 0.f;
            }
        }
    }
}

__device__ __forceinline__ void mlp_layer3_max(const float* __restrict__ W,
                                               const float* __restrict__ bias,
                                               const float* xin,
                                               float* __restrict__ out,
                                               int lane, int b, int m) {
    const int half = lane >> 4;
    const int l    = lane & 15;
    for (int mt = 0; mt < 8; ++mt) {          // O = 128
        float rm[8];
#pragma unroll
        for (int r = 0; r < 8; ++r) rm[r] = -__builtin_inff();
        for (int nt = 0; nt < 2; ++nt) {
            v8f acc = {0.f, 0.f, 0.f, 0.f, 0.f, 0.f, 0.f, 0.f};
            for (int k0 = 0; k0 < 64; k0 += 4) {
                const int kk = k0 + half * 2;
                v2f a, bb;
                a.x = W[(mt * 16 + l) * 64 + kk];
                a.y = W[(mt * 16 + l) * 64 + kk + 1];
                bb.x = xin[kk * KNB + nt * 16 + l];
                bb.y = xin[(kk + 1) * KNB + nt * 16 + l];
                acc = wmma_f32(a, bb, acc);
            }
#pragma unroll
            for (int r = 0; r < 8; ++r) {
                const int row = mt * 16 + half * 8 + r;
                float v = acc[r] + bias[row];
                v = v > 0.f ? v : 0.f;                     // ReLU before max
#pragma unroll
                for (int s = 1; s < 16; s <<= 1)           // max over 16 cols
                    v = fmaxf(v, __shfl_xor(v, s, 16));
                rm[r] = fmaxf(rm[r], v);                   // combine the 2 N-tiles
            }
        }
        if (l == 0) {                                      // lanes 0 and 16 write
#pragma unroll
            for (int r = 0; r < 8; ++r) {
                const int row = mt * 16 + half * 8 + r;
                out[((size_t)b * 128 + row) * M_CENTERS + m] = rm[r];
            }
        }
    }
}

__global__ __launch_bounds__(128) void mlp_kernel(const float* __restrict__ feats,
                                                  const float* __restrict__ coords,
                                                  const float* __restrict__ centers,
                                                  const int* __restrict__ nbidx,
                                                  const float* __restrict__ W1, const float* __restrict__ b1,
                                                  const float* __restrict__ W2, const float* __restrict__ b2,
                                                  const float* __restrict__ W3, const float* __restrict__ b3,
                                                  float* __restrict__ out) {
    __shared__ float lbuf[4][2][2048];          // 64 KB: per-wave ping-pong buffers
    const int lane = threadIdx.x & 31;
    const int w    = threadIdx.x >> 5;
    const int gw   = blockIdx.x * 4 + w;
    const int b    = gw / M_CENTERS;
    const int m    = gw % M_CENTERS;
    float* bufA = &lbuf[w][0][0];
    float* bufB = &lbuf[w][1][0];

    const float cx = centers[(size_t)b * 3 * M_CENTERS + m];
    const float cy = centers[(size_t)b * 3 * M_CENTERS + M_CENTERS + m];
    const float cz = centers[(size_t)b * 3 * M_CENTERS + 2 * M_CENTERS + m];

    // Gather: lane = neighbor k. Channels 0-2: relative coords, 3-34: features,
    // row 35: zero padding (so the K=35 GEMM tail multiplies by zero).
    const int p = nbidx[(size_t)gw * KNB + lane];
    const float* cb = coords + (size_t)b * 3 * NPTS;
    bufA[0 * KNB + lane] = cb[p] - cx;
    bufA[1 * KNB + lane] = cb[NPTS + p] - cy;
    bufA[2 * KNB + lane] = cb[2 * NPTS + p] - cz;
    const float* fb = feats + (size_t)b * CFEAT * NPTS;
    for (int c = 0; c < CFEAT; ++c)
        bufA[(3 + c) * KNB + lane] = fb[c * NPTS + p];
    bufA[35 * KNB + lane] = 0.f;

    asm volatile("s_wait_dscnt 0" ::: "memory");   // cross-lane LDS visibility (same wave)
    mlp_layer<CIN, 36, 64>(W1, b1, bufA, bufB, lane);
    asm volatile("s_wait_dscnt 0" ::: "memory");
    mlp_layer<64, 64, 64>(W2, b2, bufB, bufA, lane);
    asm volatile("s_wait_dscnt 0" ::: "memory");
    mlp_layer3_max(W3, b3, bufA, out, lane, b, m);
}

// ---------------------------------------------------------------------------
extern "C" void kernel_launch(void* const* d_in, const int* in_sizes, int n_in,
                              void* d_out, int out_size, void* d_ws, size_t ws_size,
                              hipStream_t stream) {
    (void)in_sizes; (void)n_in; (void)out_size; (void)ws_size;
    const float* feats  = (const float*)d_in[0];
    const float* coords = (const float*)d_in[1];
    const float* W1 = (const float*)d_in[2];
    const float* b1 = (const float*)d_in[3];
    const float* W2 = (const float*)d_in[4];
    const float* b2 = (const float*)d_in[5];
    const float* W3 = (const float*)d_in[6];
    const float* b3 = (const float*)d_in[7];

    float* out     = (float*)d_out;                                  // (B,128,M)
    float* centers = out + (size_t)BATCH * 128 * M_CENTERS;          // (B,3,M) tail of d_out
    int*   nbidx   = (int*)d_ws;                                     // (B,M,K) scratch

    fps_kernel<<<BATCH, 1024, 0, stream>>>(coords, centers);
    ballquery_kernel<<<(BATCH * M_CENTERS) / 8, 256, 0, stream>>>(coords, centers, nbidx);
    mlp_kernel<<<(BATCH * M_CENTERS) / 4, 128, 0, stream>>>(feats, coords, centers, nbidx,
                                                            W1, b1, W2, b2, W3, b3, out);
}